// PCAProjector_11166914970561
// MI455X (gfx1250) — compile-verified
//
#include <hip/hip_runtime.h>
#include <hip/hip_bf16.h>
#include <math.h>
#include <stdint.h>

#define NH 16
#define HS 64
#define DMODEL 1024

typedef __attribute__((ext_vector_type(2))) float v2f;
typedef __attribute__((ext_vector_type(8))) float v8f;
typedef __attribute__((ext_vector_type(4))) unsigned int v4u;
typedef __attribute__((ext_vector_type(8))) int v8i;
typedef __attribute__((ext_vector_type(4))) int v4i;

// ---------------------------------------------------------------------------
// TDM helper: 2D tile load, 64 rows x 64 f32, row stride DMODEL elements.
// D# group0: count=1 | lds_addr | global_addr | type=2
// D# group1: data_size=4B, tensor_dims large (no OOB), tile 64x64, stride0=1024
// Caller must be a single wave (TDM issues once per executing wave).
// ---------------------------------------------------------------------------
static __device__ __forceinline__ void tdm_load_tile64(const float* gptr,
                                                       unsigned lds_off) {
  const unsigned long long ga = (unsigned long long)(size_t)gptr;
  v4u g0;
  g0[0] = 1u;                                   // count=1, user mode
  g0[1] = lds_off;                              // LDS byte address
  g0[2] = (unsigned)(ga & 0xFFFFFFFFu);         // global_addr[31:0]
  g0[3] = (unsigned)((ga >> 32) & 0x01FFFFFFu)  // global_addr[56:32]
          | 0x80000000u;                        // type=2 ("image")
  const unsigned TD0 = 0x00100000u;             // big tensor dims: disable OOB
  const unsigned TD1 = 0x00100000u;
  v8i g1;
  g1[0] = (int)(2u << 16);                      // data_size=2 (4 bytes)
  g1[1] = (int)((TD0 & 0xFFFFu) << 16);         // tensor_dim0[15:0]
  g1[2] = (int)((TD0 >> 16) | ((TD1 & 0xFFFFu) << 16));
  g1[3] = (int)((TD1 >> 16) | (64u << 16));     // tile_dim0 = 64
  g1[4] = (int)64;                              // tile_dim1 = 64, tile_dim2=0
  g1[5] = (int)DMODEL;                          // tensor_dim0_stride = 1024
  g1[6] = 0;
  g1[7] = 0;
  const v4i zz4 = {0, 0, 0, 0};
  const v8i zz8 = {0, 0, 0, 0, 0, 0, 0, 0};
  __builtin_amdgcn_tensor_load_to_lds(g0, g1, zz4, zz4, zz8, 0);
}

// ---------------------------------------------------------------------------
// Kernel 1: partial Gram  G_part = Y_chunk Y_chunk^T  plus partial row sums.
//   Y[i, s] = x[b, s, h*64 + i]
// grid = B*NH*split blocks, block = 512 threads (16 waves, 16x16 tile each).
// TDM double-buffered global->LDS pipeline; V_WMMA_F32_16X16X4_F32 compute.
// ---------------------------------------------------------------------------
__global__ __launch_bounds__(512) void pca_cov_partial_kernel(
    const float* __restrict__ x, float* __restrict__ gpart,
    float* __restrict__ rspart, int S, int split) {
  const int blk  = blockIdx.x;           // bh*split + part
  const int bh   = blk / split;
  const int part = blk % split;
  const int b    = bh / NH;
  const int h    = bh % NH;
  const int tid  = threadIdx.x;
  const int wave = tid >> 5;
  const int lane = tid & 31;
  const int half = lane >> 4;
  const int l15  = lane & 15;
  const int ti   = (wave >> 2) * 16;
  const int tj   = (wave & 3) * 16;
  const int li   = tid & 63;             // i for rowsum accumulation
  const int ls   = tid >> 6;             // s sub-slot 0..7

  __shared__ float tile[2][64][64];      // double buffer (TDM destination)
  __shared__ float rsum_part[8][64];

  const int Sw  = S / split;             // rows handled by this WG
  const int nch = Sw >> 6;               // 64-row chunks
  const float* xb =
      x + ((size_t)b * (size_t)S + (size_t)part * Sw) * DMODEL + h * HS;

  unsigned ldsb[2];
  ldsb[0] = (unsigned)(size_t)&tile[0][0][0];
  ldsb[1] = (unsigned)(size_t)&tile[1][0][0];

  v8f acc = {0.f, 0.f, 0.f, 0.f, 0.f, 0.f, 0.f, 0.f};
  float rs_acc = 0.f;

  if (wave == 0) tdm_load_tile64(xb, ldsb[0]);  // prefetch chunk 0

  for (int i = 0; i < nch; ++i) {
    const int buf = i & 1;
    if (wave == 0) {
      if (i + 1 < nch) {
        tdm_load_tile64(xb + (size_t)(i + 1) * 64 * DMODEL, ldsb[(i + 1) & 1]);
        __builtin_amdgcn_s_wait_tensorcnt(1);   // chunk i done (in-order)
      } else {
        __builtin_amdgcn_s_wait_tensorcnt(0);
      }
    }
    __syncthreads();                            // chunk i visible to all waves

    // row-sum partials out of LDS (i fixed per thread -> conflict-free)
    #pragma unroll
    for (int j = 0; j < 8; ++j) rs_acc += tile[buf][ls + j * 8][li];

    // Gram accumulation over K=64 in fp32 WMMA 16x16x4 steps
    #pragma unroll
    for (int kk = 0; kk < 16; ++kk) {
      const int kb = kk * 4 + half * 2;
      v2f a, bf;
      a.x  = tile[buf][kb][ti + l15];           // A[m][k] = Y[ti+m][kb]
      a.y  = tile[buf][kb + 1][ti + l15];
      bf.x = tile[buf][kb][tj + l15];           // B[k][n] = Y[tj+n][kb]
      bf.y = tile[buf][kb + 1][tj + l15];
      acc = __builtin_amdgcn_wmma_f32_16x16x4_f32(
          false, a, false, bf, (short)0, acc, false, false);
    }
    __syncthreads();   // all readers done before TDM overwrites this buffer
  }

  // reduce row-sum partials (8 slots per i)
  rsum_part[ls][li] = rs_acc;
  __syncthreads();
  if (tid < 64) {
    float r = 0.f;
    #pragma unroll
    for (int j = 0; j < 8; ++j) r += rsum_part[j][tid];
    rspart[(size_t)blk * 64 + tid] = r;
  }

  // store raw partial Gram (correction applied in reduce kernel)
  float* gp = gpart + (size_t)blk * 4096;
  #pragma unroll
  for (int r = 0; r < 8; ++r) {
    const int m = ti + r + half * 8;            // C/D layout: VGPR r -> M
    const int n = tj + l15;
    gp[m * 64 + n] = acc[r];
  }
}

// ---------------------------------------------------------------------------
// Kernel 1b: deterministic fixed-order reduction of split partials +
// mean-centering correction:  cov = sum_p G_p - rowsum rowsum^T / S
// ---------------------------------------------------------------------------
__global__ __launch_bounds__(256) void pca_cov_reduce_kernel(
    const float* __restrict__ gpart, const float* __restrict__ rspart,
    float* __restrict__ cov, int split, int S) {
  const int bh  = blockIdx.x;
  const int tid = threadIdx.x;
  __shared__ float rowsum[64];
  if (tid < 64) {
    float r = 0.f;
    for (int p = 0; p < split; ++p)
      r += rspart[((size_t)bh * split + p) * 64 + tid];
    rowsum[tid] = r;
  }
  __syncthreads();
  const float invS = 1.0f / (float)S;
  for (int e = tid; e < 4096; e += 256) {
    float g = 0.f;
    for (int p = 0; p < split; ++p)
      g += gpart[((size_t)bh * split + p) * 4096 + e];
    const int m = e >> 6, n = e & 63;
    cov[(size_t)bh * 4096 + e] = g - rowsum[m] * rowsum[n] * invS;
  }
}

// ---------------------------------------------------------------------------
// Kernel 2: symmetric eigensolve, parallel cyclic two-sided Jacobi.
// One workgroup (256 threads) per 64x64 matrix; tournament pairing, 32
// disjoint rotations/stage, 63 stages/sweep. Ascending sort (eigh order).
// ---------------------------------------------------------------------------
__global__ __launch_bounds__(256) void pca_jacobi_kernel(
    const float* __restrict__ cov, float* __restrict__ vals,
    float* __restrict__ vecs, int nsweep) {
  const int bh  = blockIdx.x;
  const int tid = threadIdx.x;

  __shared__ float A[64][65];
  __shared__ float V[64][65];
  __shared__ float cs[32], sn[32];
  __shared__ int   perm[64];
  __shared__ float dvals[64];
  __shared__ int   rnk[64];

  const float* covb = cov + (size_t)bh * 4096;
  for (int e = tid; e < 4096; e += 256) {
    const int r = e >> 6, c = e & 63;
    A[r][c] = covb[e];
    V[r][c] = (r == c) ? 1.f : 0.f;
  }
  if (tid < 64) perm[tid] = tid;
  __syncthreads();

  for (int sw = 0; sw < nsweep; ++sw) {
    for (int stage = 0; stage < 63; ++stage) {
      if (tid < 32) {
        int p = perm[tid], q = perm[63 - tid];
        if (p > q) { int t = p; p = q; q = t; }
        const float app = A[p][p], aqq = A[q][q], apq = A[p][q];
        float c = 1.f, s = 0.f;
        if (fabsf(apq) > 1e-30f) {
          const float tau = (aqq - app) / (2.f * apq);
          const float t =
              copysignf(1.f, tau) / (fabsf(tau) + sqrtf(1.f + tau * tau));
          c = rsqrtf(1.f + t * t);
          s = t * c;
        }
        cs[tid] = c; sn[tid] = s;
      }
      __syncthreads();
      // column rotations: A <- A*J, V <- V*J (disjoint pairs)
      for (int w = tid; w < 64 * 32; w += 256) {
        const int r = w >> 5, k = w & 31;
        int p = perm[k], q = perm[63 - k];
        if (p > q) { int t = p; p = q; q = t; }
        const float c = cs[k], s = sn[k];
        const float xp = A[r][p], xq = A[r][q];
        A[r][p] = c * xp - s * xq;
        A[r][q] = s * xp + c * xq;
        const float vp = V[r][p], vq = V[r][q];
        V[r][p] = c * vp - s * vq;
        V[r][q] = s * vp + c * vq;
      }
      __syncthreads();
      // row rotations: A <- J^T * A
      for (int w = tid; w < 64 * 32; w += 256) {
        const int cI = w >> 5, k = w & 31;
        int p = perm[k], q = perm[63 - k];
        if (p > q) { int t = p; p = q; q = t; }
        const float c = cs[k], s = sn[k];
        const float xp = A[p][cI], xq = A[q][cI];
        A[p][cI] = c * xp - s * xq;
        A[q][cI] = s * xp + c * xq;
      }
      __syncthreads();
      if (tid == 0) {                     // rotate tournament schedule
        const int last = perm[63];
        for (int i = 63; i > 1; --i) perm[i] = perm[i - 1];
        perm[1] = last;
      }
      __syncthreads();
    }
  }

  if (tid < 64) dvals[tid] = A[tid][tid];
  __syncthreads();
  if (tid < 64) {
    const float v = dvals[tid];
    int rk = 0;
    for (int j = 0; j < 64; ++j) {
      const float u = dvals[j];
      if (u < v || (u == v && j < tid)) ++rk;
    }
    rnk[tid] = rk;
  }
  __syncthreads();

  float* valsb = vals + bh * 64;
  float* vecsb = vecs + (size_t)bh * 4096;
  if (tid < 64) valsb[rnk[tid]] = dvals[tid];
  for (int e = tid; e < 4096; e += 256) {
    const int r = e >> 6, c = e & 63;
    vecsb[r * 64 + rnk[c]] = V[r][c];
  }
}

// ---------------------------------------------------------------------------
// Kernel 3: out2 = (vecs^T diag(vals)) @ W + b, scattered head-interleaved:
//   z[b, i, h*64 + j] = out2[i][j].  fp32 WMMA, K=64.
// ---------------------------------------------------------------------------
__global__ __launch_bounds__(512) void pca_project_kernel(
    const float* __restrict__ vals, const float* __restrict__ vecs,
    const float* __restrict__ W, const float* __restrict__ bvec,
    float* __restrict__ z) {
  const int bh   = blockIdx.x;
  const int b    = bh / NH;
  const int h    = bh % NH;
  const int tid  = threadIdx.x;
  const int wave = tid >> 5;
  const int lane = tid & 31;
  const int half = lane >> 4;
  const int l15  = lane & 15;
  const int ti   = (wave >> 2) * 16;
  const int tj   = (wave & 3) * 16;

  __shared__ float Am[64][68];           // A[i][k] = vals[k]*vecs[k][i]
  __shared__ float Wm[64][68];           // W[k][j]

  const float* vb = vecs + (size_t)bh * 4096;
  const float* lv = vals + bh * 64;
  for (int e = tid; e < 4096; e += 512) {
    const int k = e >> 6, i = e & 63;
    Am[i][k] = lv[k] * vb[k * 64 + i];
    Wm[k][i] = W[e];
  }
  __syncthreads();

  v8f acc = {0.f, 0.f, 0.f, 0.f, 0.f, 0.f, 0.f, 0.f};
  #pragma unroll
  for (int kk = 0; kk < 16; ++kk) {
    const int kb = kk * 4 + half * 2;
    v2f a, bf;
    a.x  = Am[ti + l15][kb];
    a.y  = Am[ti + l15][kb + 1];
    bf.x = Wm[kb][tj + l15];
    bf.y = Wm[kb + 1][tj + l15];
    acc = __builtin_amdgcn_wmma_f32_16x16x4_f32(
        false, a, false, bf, (short)0, acc, false, false);
  }

  float* zb = z + ((size_t)b * 64) * DMODEL + h * HS;
  #pragma unroll
  for (int r = 0; r < 8; ++r) {
    const int m = ti + r + half * 8;
    const int n = tj + l15;
    zb[(size_t)m * DMODEL + n] = acc[r] + bvec[n];
  }
}

// ---------------------------------------------------------------------------
// Kernel 4: final projection  out[b,i] = sum_d z[b,i,d]*pw[d] + pb
// ---------------------------------------------------------------------------
__global__ __launch_bounds__(256) void pca_final_kernel(
    const float* __restrict__ z, const float* __restrict__ pw,
    const float* __restrict__ pb, float* __restrict__ out) {
  const int row = blockIdx.x;
  const int tid = threadIdx.x;
  const float* zr = z + (size_t)row * DMODEL;
  float acc = 0.f;
  for (int d = tid; d < DMODEL; d += 256) acc += zr[d] * pw[d];
  __shared__ float red[256];
  red[tid] = acc;
  __syncthreads();
  for (int off = 128; off > 0; off >>= 1) {
    if (tid < off) red[tid] += red[tid + off];
    __syncthreads();
  }
  if (tid == 0) out[row] = red[0] + pb[0];
}

// ---------------------------------------------------------------------------
extern "C" void kernel_launch(void* const* d_in, const int* in_sizes, int n_in,
                              void* d_out, int out_size, void* d_ws,
                              size_t ws_size, hipStream_t stream) {
  (void)n_in; (void)out_size; (void)ws_size;
  const float* x    = (const float*)d_in[0];  // [B, 4096, 1024]
  const float* dffW = (const float*)d_in[1];  // [3, 64, 64]
  const float* dffb = (const float*)d_in[2];  // [3, 64]
  const float* pw   = (const float*)d_in[3];  // [1024, 1]
  const float* pb   = (const float*)d_in[4];  // [1]
  float* out = (float*)d_out;

  const int S1    = 4096;
  const int B     = in_sizes[0] / (S1 * DMODEL);  // = 8
  const int G     = B * NH;                       // 128 matrices
  const int SPLIT = 4;                            // layer-1 S split

  // workspace layout (floats)
  float* ws    = (float*)d_ws;
  float* gpart = ws;                                   // G*SPLIT*4096
  float* rsprt = gpart + (size_t)G * SPLIT * 4096;     // G*SPLIT*64
  float* cov   = rsprt + (size_t)G * SPLIT * 64;       // G*4096
  float* vals  = cov + (size_t)G * 4096;               // G*64
  float* vecs  = vals + (size_t)G * 64;                // G*4096
  float* z1    = vecs + (size_t)G * 4096;              // B*64*1024
  float* z2    = z1 + (size_t)B * 64 * DMODEL;         // B*64*1024

  const int NSWEEP = 8;

  // ---- layer 1 (S = 4096, split over S for occupancy) ----
  pca_cov_partial_kernel<<<G * SPLIT, 512, 0, stream>>>(x, gpart, rsprt, S1,
                                                        SPLIT);
  pca_cov_reduce_kernel<<<G, 256, 0, stream>>>(gpart, rsprt, cov, SPLIT, S1);
  pca_jacobi_kernel<<<G, 256, 0, stream>>>(cov, vals, vecs, NSWEEP);
  pca_project_kernel<<<G, 512, 0, stream>>>(vals, vecs, dffW, dffb, z1);

  // ---- layer 2 (S = 64) ----
  pca_cov_partial_kernel<<<G, 512, 0, stream>>>(z1, gpart, rsprt, 64, 1);
  pca_cov_reduce_kernel<<<G, 256, 0, stream>>>(gpart, rsprt, cov, 1, 64);
  pca_jacobi_kernel<<<G, 256, 0, stream>>>(cov, vals, vecs, NSWEEP);
  pca_project_kernel<<<G, 512, 0, stream>>>(vals, vecs, dffW + 4096,
                                            dffb + 64, z2);

  // ---- layer 3 (S = 64) ----
  pca_cov_partial_kernel<<<G, 512, 0, stream>>>(z2, gpart, rsprt, 64, 1);
  pca_cov_reduce_kernel<<<G, 256, 0, stream>>>(gpart, rsprt, cov, 1, 64);
  pca_jacobi_kernel<<<G, 256, 0, stream>>>(cov, vals, vecs, NSWEEP);
  pca_project_kernel<<<G, 512, 0, stream>>>(vals, vecs, dffW + 2 * 4096,
                                            dffb + 128, z1);

  // ---- final 1024 -> 1 projection ----
  pca_final_kernel<<<B * 64, 256, 0, stream>>>(z1, pw, pb, out);
}